// EctLayer_1769526526456
// MI455X (gfx1250) — compile-verified
//
#include <hip/hip_runtime.h>
#include <math.h>

// ECT layer: nh = x@v  ->  sigmoid(lin[s]-nh[t])  ->  segment_sum over sorted batch.
// Restructured: sigmoid(lin[s]-nh) = 1/(1 + exp(nh)*exp(-lin[s]))
//   phase 1: WMMA f32 16x16x4 projection (F=3 padded to K=4) + exp(nh) -> LDS
//   phase 2: per-(s,t) thread walks chunk: acc += rcp(fma(E, d_s, 1)), flush
//            to global with atomicAdd at segment boundaries.

typedef __attribute__((ext_vector_type(2))) float v2f;
typedef __attribute__((ext_vector_type(8))) float v8f;

#define CHUNK 256
#define T_DIM 32
#define S_DIM 32
#define LDS_STRIDE 33   // pad to avoid 2-way bank conflicts on WMMA D store

__global__ void ect_zero_kernel(float* __restrict__ out, int n) {
    int i = blockIdx.x * blockDim.x + threadIdx.x;
    if (i < n) out[i] = 0.0f;
}

__global__ __launch_bounds__(1024) void ect_kernel(
    const float* __restrict__ x,      // [N,3]
    const int*   __restrict__ batch,  // [N], sorted
    const float* __restrict__ v,      // [3,32]
    const float* __restrict__ lin,    // [32]
    float*       __restrict__ out,    // [128,32,32]
    int N)
{
    __shared__ float E[CHUNK * LDS_STRIDE]; // exp(nh[p][t])
    __shared__ int   bat[CHUNK];

    const int tid  = threadIdx.x;
    const int lane = tid & 31;
    const int wave = tid >> 5;
    const long long chunk_start = (long long)blockIdx.x * CHUNK;

    // stage batch ids for this chunk
    if (tid < CHUNK) {
        long long p = chunk_start + tid;
        bat[tid] = (p < N) ? batch[p] : -1;
    }

    // ---- Phase 1: projection via V_WMMA_F32_16X16X4_F32, then exp -> LDS ----
    {
        const int ptile = wave >> 1;      // 0..15 : which 16-point tile
        const int ttile = wave & 1;       // 0..1  : which 16-direction tile
        const int hi    = lane >> 4;      // lanes 16-31 carry K=2,3
        const int l15   = lane & 15;

        long long p = chunk_start + (long long)ptile * 16 + l15;
        if (p >= N) p = N - 1;            // clamp; tail masked in phase 2

        // A (16x4, f32): lanes 0-15 -> K=0,1 ; lanes 16-31 -> K=2,3 (K=3 pad=0)
        v2f a;
        a.x = x[p * 3 + (hi ? 2 : 0)];
        a.y = hi ? 0.0f : x[p * 3 + 1];

        // B (4x16, f32): column n = ttile*16 + l15, mirrored K striping
        const int t = ttile * 16 + l15;
        v2f b;
        b.x = hi ? v[2 * T_DIM + t] : v[0 * T_DIM + t];
        b.y = hi ? 0.0f : v[1 * T_DIM + t];

        v8f c = {0.f, 0.f, 0.f, 0.f, 0.f, 0.f, 0.f, 0.f};
        c = __builtin_amdgcn_wmma_f32_16x16x4_f32(
                /*neg_a=*/false, a, /*neg_b=*/false, b,
                /*c_mod=*/(short)0, c, /*reuse_a=*/false, /*reuse_b=*/false);

        // D layout: VGPR r holds M=r (lanes 0-15) / M=r+8 (lanes 16-31), N=lane%16
        const int rowbase = ptile * 16 + hi * 8;
        #pragma unroll
        for (int r = 0; r < 8; ++r) {
            E[(rowbase + r) * LDS_STRIDE + t] = __expf(c[r]);
        }
    }
    __syncthreads();

    // ---- Phase 2: per-(s,t) accumulation with segment-boundary flushes ----
    const int t = tid & 31;           // direction
    const int s = tid >> 5;           // threshold; tid == s*32 + t
    const float ds = __expf(-lin[s]); // exp(-lin[s])

    float acc = 0.0f;
    int cur = bat[0];
    #pragma unroll 4
    for (int p = 0; p < CHUNK; ++p) {
        if (chunk_start + p >= N) break;
        int b = bat[p];
        if (b != cur) {
            atomicAdd(&out[cur * (S_DIM * T_DIM) + tid], acc);
            acc = 0.0f;
            cur = b;
        }
        float e = E[p * LDS_STRIDE + t];
        acc += __builtin_amdgcn_rcpf(fmaf(e, ds, 1.0f)); // sigmoid value
    }
    if (cur >= 0) {
        atomicAdd(&out[cur * (S_DIM * T_DIM) + tid], acc);
    }
}

extern "C" void kernel_launch(void* const* d_in, const int* in_sizes, int n_in,
                              void* d_out, int out_size, void* d_ws, size_t ws_size,
                              hipStream_t stream) {
    const float* x     = (const float*)d_in[0];
    const int*   batch = (const int*)  d_in[1];
    const float* v     = (const float*)d_in[2];
    const float* lin   = (const float*)d_in[3];
    float*       out   = (float*)d_out;

    const int N = in_sizes[1]; // batch length == number of points

    // zero the output (harness poisons it; we accumulate with atomics)
    ect_zero_kernel<<<(out_size + 255) / 256, 256, 0, stream>>>(out, out_size);

    const int nblocks = (N + CHUNK - 1) / CHUNK;
    ect_kernel<<<nblocks, 1024, 0, stream>>>(x, batch, v, lin, out, N);
}